// RateDistortionLossV2_1898375545142
// MI455X (gfx1250) — compile-verified
//
#include <hip/hip_runtime.h>
#include <math.h>

typedef __attribute__((ext_vector_type(2))) float v2f;
typedef __attribute__((ext_vector_type(8))) float v8f;

#define BLOCK 256
#define NB_MSE 1024
#define NB_RY  2048
#define NB_RZ  128

// ---------------------------------------------------------------------------
// Lane-XOR add via ds_swizzle (group-of-32 mode: offset = xor<<10 | and=0x1F),
// one DS instruction per step instead of HIP's bounds-checked ds_bpermute.
// ---------------------------------------------------------------------------
template <int PATTERN>
__device__ __forceinline__ float swz_add(float v) {
#if __has_builtin(__builtin_amdgcn_ds_swizzle)
    int t = __builtin_amdgcn_ds_swizzle(__float_as_int(v), PATTERN);
    return v + __int_as_float(t);
#else
    return v + __shfl_xor(v, (PATTERN >> 10), 32);
#endif
}

// ---------------------------------------------------------------------------
// Wave32 sum reduction.
// CDNA5 path: one exact fp32 WMMA (V_WMMA_F32_16X16X4_F32). A = ones(16x4)
// (fills every A slot with 1.0 regardless of lane->(M,K) mapping). B holds
// each lane's value in exactly one (K,N) slot (b[0]=v, b[1]=0), so
// D[m,n] = colsum(n) for every m and c[0] = colsum(lane&15) in every lane.
// Four XOR-swizzle adds over the 16 columns finish the exact wave sum.
// ---------------------------------------------------------------------------
__device__ __forceinline__ float wave_reduce_sum(float v) {
#if __has_builtin(__builtin_amdgcn_wmma_f32_16x16x4_f32)
    v2f a; a[0] = 1.0f; a[1] = 1.0f;   // A = ones 16x4
    v2f b; b[0] = v;    b[1] = 0.0f;   // one live slot per lane
    v8f c = {};
    c = __builtin_amdgcn_wmma_f32_16x16x4_f32(
        /*neg_a=*/false, a, /*neg_b=*/false, b,
        /*c_mod=*/(short)0, c, /*reuse_a=*/false, /*reuse_b=*/false);
    float s = c[0];
    s = swz_add<0x041F>(s);   // xor 1
    s = swz_add<0x081F>(s);   // xor 2
    s = swz_add<0x101F>(s);   // xor 4
    s = swz_add<0x201F>(s);   // xor 8
    return s;                 // full wave sum in every lane
#else
    for (int m = 16; m >= 1; m >>= 1) v += __shfl_xor(v, m, 32);
    return v;
#endif
}

// Block reduction: wave WMMA-reduce, then sum the 8 wave results via LDS.
// Result valid on threadIdx.x == 0.
__device__ __forceinline__ float block_reduce_sum(float v) {
    __shared__ float smem[BLOCK / 32];
    v = wave_reduce_sum(v);
    const int lane = threadIdx.x & 31;
    const int wid  = threadIdx.x >> 5;
    if (lane == 0) smem[wid] = v;
    __syncthreads();
    float r = 0.0f;
    if (threadIdx.x == 0) {
#pragma unroll
        for (int i = 0; i < BLOCK / 32; ++i) r += smem[i];
    }
    __syncthreads();   // allow safe reuse of smem by a following call
    return r;
}

// ---------------------------------------------------------------------------
// Math helpers (v_exp_f32 / v_log_f32 backed intrinsics; erfc stays accurate)
// ---------------------------------------------------------------------------
__device__ __forceinline__ float std_cdf(float t) {
    // 0.5 * erfc(-t / sqrt(2))
    return 0.5f * erfcf(-0.70710678118654752f * t);
}

__device__ __forceinline__ float log_pmf_component(float y, float mean, float raw_scale) {
    float scale = fmaxf(__expf(raw_scale), 1e-5f);
#if __has_builtin(__builtin_amdgcn_rcpf)
    float inv = __builtin_amdgcn_rcpf(scale);
#else
    float inv = 1.0f / scale;
#endif
    float up  = (y + 0.5f - mean) * inv;
    float lo  = (y - 0.5f - mean) * inv;
    float pmf = std_cdf(up) - std_cdf(lo);
    return __logf(fmaxf(pmf, 1e-6f));
}

__device__ __forceinline__ float gmm_rate(float yv,
                                          float w0, float w1, float w2,
                                          float m0, float m1, float m2,
                                          float s0, float s1, float s2) {
    // log-softmax over the 3 mixture logits
    float wm  = fmaxf(fmaxf(w0, w1), w2);
    float lse = wm + __logf(__expf(w0 - wm) + __expf(w1 - wm) + __expf(w2 - wm));

    float t0 = (w0 - lse) + log_pmf_component(yv, m0, s0);
    float t1 = (w1 - lse) + log_pmf_component(yv, m1, s1);
    float t2 = (w2 - lse) + log_pmf_component(yv, m2, s2);

    // logsumexp over mixture components
    float tm  = fmaxf(fmaxf(t0, t1), t2);
    float lpt = tm + __logf(__expf(t0 - tm) + __expf(t1 - tm) + __expf(t2 - tm));

    return -lpt * 1.44269504088896340736f;   // / ln(2)
}

// ---------------------------------------------------------------------------
// Kernel 1: sum of squared differences (MSE numerator), float4 streaming.
// ---------------------------------------------------------------------------
__global__ void mse_partial_kernel(const float4* __restrict__ x,
                                   const float4* __restrict__ xh,
                                   float* __restrict__ part, int n4) {
    float acc = 0.0f;
    const int stride = gridDim.x * blockDim.x;
    for (int i = blockIdx.x * blockDim.x + threadIdx.x; i < n4; i += stride) {
        float4 a = x[i];
        float4 b = xh[i];
        float d0 = b.x - a.x, d1 = b.y - a.y, d2 = b.z - a.z, d3 = b.w - a.w;
        acc = fmaf(d0, d0, acc);
        acc = fmaf(d1, d1, acc);
        acc = fmaf(d2, d2, acc);
        acc = fmaf(d3, d3, acc);
    }
    float r = block_reduce_sum(acc);
    if (threadIdx.x == 0) part[blockIdx.x] = r;
}

// ---------------------------------------------------------------------------
// Kernel 2: GMM latent rate, float4-vectorized (HW=2304 % 4 == 0, so a group
// of 4 consecutive hw never crosses a channel). Per group: 9 param float4
// loads + 1 y float4 load, all b128 and fully coalesced across the wave.
// params channel = c*9 + j*3 + k (j = logit/mean/log-scale, k = mixture).
// ---------------------------------------------------------------------------
__global__ void rate_y_partial_kernel(const float4* __restrict__ y4,
                                      const float4* __restrict__ params4,
                                      float* __restrict__ part, int n4) {
    const int HW4  = (48 * 48) / 4;   // 576
    const int CHW4 = 192 * HW4;       // 110592
    float acc = 0.0f;
    const int stride = gridDim.x * blockDim.x;
    for (int i = blockIdx.x * blockDim.x + threadIdx.x; i < n4; i += stride) {
        int b   = i / CHW4;
        int rem = i - b * CHW4;
        int c   = rem / HW4;
        int hw4 = rem - c * HW4;
        const float4* p = params4 + (((size_t)b * 1728 + (size_t)c * 9) * HW4 + hw4);

        float4 yv = y4[i];
        float4 W0 = p[0 * HW4], W1 = p[1 * HW4], W2 = p[2 * HW4];
        float4 M0 = p[3 * HW4], M1 = p[4 * HW4], M2 = p[5 * HW4];
        float4 S0 = p[6 * HW4], S1 = p[7 * HW4], S2 = p[8 * HW4];

        acc += gmm_rate(yv.x, W0.x, W1.x, W2.x, M0.x, M1.x, M2.x, S0.x, S1.x, S2.x);
        acc += gmm_rate(yv.y, W0.y, W1.y, W2.y, M0.y, M1.y, M2.y, S0.y, S1.y, S2.y);
        acc += gmm_rate(yv.z, W0.z, W1.z, W2.z, M0.z, M1.z, M2.z, S0.z, S1.z, S2.z);
        acc += gmm_rate(yv.w, W0.w, W1.w, W2.w, M0.w, M1.w, M2.w, S0.w, S1.w, S2.w);
    }
    float r = block_reduce_sum(acc);
    if (threadIdx.x == 0) part[blockIdx.x] = r;
}

// ---------------------------------------------------------------------------
// Kernel 3: hyperlatent rate, float4 streaming.
// ---------------------------------------------------------------------------
__global__ void rate_z_partial_kernel(const float4* __restrict__ z4,
                                      float* __restrict__ part, int n4) {
    float acc = 0.0f;
    const int stride = gridDim.x * blockDim.x;
    for (int i = blockIdx.x * blockDim.x + threadIdx.x; i < n4; i += stride) {
        float4 zv = z4[i];
#pragma unroll
        for (int j = 0; j < 4; ++j) {
            float z   = (&zv.x)[j];
            float pmf = fmaxf(std_cdf(z + 0.5f) - std_cdf(z - 0.5f), 1e-6f);
            acc += -__log2f(pmf);
        }
    }
    float r = block_reduce_sum(acc);
    if (threadIdx.x == 0) part[blockIdx.x] = r;
}

// ---------------------------------------------------------------------------
// Kernel 4: deterministic final reduction of all block partials + scalar math.
// ---------------------------------------------------------------------------
__global__ void finalize_kernel(const float* __restrict__ ws,
                                float* __restrict__ out, int n_x) {
    float a = 0.0f, b = 0.0f, c = 0.0f;
    for (int i = threadIdx.x; i < NB_MSE; i += BLOCK) a += ws[i];
    for (int i = threadIdx.x; i < NB_RY;  i += BLOCK) b += ws[NB_MSE + i];
    for (int i = threadIdx.x; i < NB_RZ;  i += BLOCK) c += ws[NB_MSE + NB_RY + i];

    a = block_reduce_sum(a);
    b = block_reduce_sum(b);
    c = block_reduce_sum(c);

    if (threadIdx.x == 0) {
        const float B_batch = 16.0f;
        float distortion = a / (float)n_x;                 // mean squared error
        float rate_y     = b / B_batch;                    // per-batch mean of sums
        float rate_z     = c / B_batch;
        float num_pixels = (float)(n_x / 3);               // B * img * img
        float bpp        = (rate_y + rate_z) / num_pixels;
        float loss       = 0.01f * 255.0f * 255.0f * distortion + bpp;
        out[0] = loss;
        out[1] = distortion;
        out[2] = bpp;
    }
}

// ---------------------------------------------------------------------------
extern "C" void kernel_launch(void* const* d_in, const int* in_sizes, int n_in,
                              void* d_out, int out_size, void* d_ws, size_t ws_size,
                              hipStream_t stream) {
    (void)n_in; (void)out_size; (void)ws_size;

    const float* x      = (const float*)d_in[0];
    const float* x_hat  = (const float*)d_in[1];
    const float* params = (const float*)d_in[2];
    const float* y_hat  = (const float*)d_in[3];
    const float* z_hat  = (const float*)d_in[4];
    float*       out    = (float*)d_out;
    float*       ws     = (float*)d_ws;

    const int n_x = in_sizes[0];   // 16*3*768*768 = 28,311,552 (divisible by 4)
    const int n_y = in_sizes[3];   // 16*192*48*48 = 7,077,888
    const int n_z = in_sizes[4];   // 16*192*12*12 = 442,368

    mse_partial_kernel<<<NB_MSE, BLOCK, 0, stream>>>(
        (const float4*)x, (const float4*)x_hat, ws, n_x / 4);
    rate_y_partial_kernel<<<NB_RY, BLOCK, 0, stream>>>(
        (const float4*)y_hat, (const float4*)params, ws + NB_MSE, n_y / 4);
    rate_z_partial_kernel<<<NB_RZ, BLOCK, 0, stream>>>(
        (const float4*)z_hat, ws + NB_MSE + NB_RY, n_z / 4);
    finalize_kernel<<<1, BLOCK, 0, stream>>>(ws, out, n_x);
}